// AttentionConcat_64123861729624
// MI455X (gfx1250) — compile-verified
//
#include <hip/hip_runtime.h>

#define B_  4
#define TQ_ 512
#define TP_ 512
#define D_  256

typedef __attribute__((ext_vector_type(2))) float v2f;
typedef __attribute__((ext_vector_type(8))) float v8f;

// ---------------------------------------------------------------------------
// Generic fp32 GEMM via V_WMMA_F32_16X16X4_F32, one 16x16 C tile per wave32.
// C[b] = A[b] (MxK, row-major, ld=K) * Bm[b] (KxN, row-major, ld=N)
// A fragment (16x4 f32): lanes 0-15 hold M=0..15 / K=k,k+1 ; lanes 16-31 K=k+2,k+3
// B fragment (4x16 f32): lanes 0-15 hold N=0..15 rows K=k,k+1 ; lanes 16-31 K=k+2,k+3
// C/D (16x16 f32): VGPR r -> M = r (lanes 0-15) / M = 8+r (lanes 16-31)
// ---------------------------------------------------------------------------
__global__ __launch_bounds__(256) void gemm_f32_wmma(
    const float* __restrict__ A, const float* __restrict__ Bm,
    float* __restrict__ C, int M, int N, int K,
    long strideA, long strideB, long strideC)
{
    const int wavesPerBlk = blockDim.x >> 5;
    const int wave = blockIdx.x * wavesPerBlk + (threadIdx.x >> 5);
    const int lane = threadIdx.x & 31;
    const int tilesN = N >> 4;
    const int tM = (wave / tilesN) << 4;
    const int tN = (wave % tilesN) << 4;
    if (tM >= M) return;                       // wave-uniform, EXEC stays all-1s

    const float* Ab = A  + (long)blockIdx.z * strideA;
    const float* Bb = Bm + (long)blockIdx.z * strideB;
    float*       Cb = C  + (long)blockIdx.z * strideC;

    const int l15  = lane & 15;
    const int koff = (lane >> 4) << 1;         // 0 for lanes 0-15, 2 for 16-31

    const float* arow = Ab + (long)(tM + l15) * K + koff;
    const float* bcol = Bb + (long)koff * N + (tN + l15);

    v8f c = {};
    for (int k = 0; k < K; k += 4) {
        v2f a, b;
        a.x = arow[k];
        a.y = arow[k + 1];
        b.x = bcol[(long)k * N];
        b.y = bcol[(long)(k + 1) * N];
        c = __builtin_amdgcn_wmma_f32_16x16x4_f32(
                false, a, false, b, (short)0, c, false, false);
    }

    const int crow = tM + ((lane >> 4) << 3);  // +8 for upper half-wave
    const int ccol = tN + l15;
#pragma unroll
    for (int r = 0; r < 8; ++r)
        Cb[(long)(crow + r) * N + ccol] = c[r];
}

// ---------------------------------------------------------------------------
// scores[b,p,q] = sum_e vc[e] * tanh(pp[b,p,e] + pq[b,q,e])
// 64p x 16q tile per 256-thread block; each thread owns 4 p-rows for one q
// (amortizes pq/vc LDS reads 4x). All LDS reads are b128:
//   pp : p-major [64][256]      -> 2 distinct addrs/wave -> LDS broadcast
//   pq : qi-major [16][260]     -> padded row stride; lane qi reads float4 at
//        qi*260+e -> banks (qi*4+e)%64 distinct across half-wave, conflict-free
//   vc : [256]                  -> broadcast
// tanh: native gfx1250 v_tanh_f32, 4 per asm block so every trans result has
// >=3 independent instructions before first use (TRANS hazard rule).
// Dynamic LDS: 83,200 B (gfx1250 allows up to 320 KB/WG; ~3 blocks/WGP).
// ---------------------------------------------------------------------------
#define PP_ROWS   64
#define PQ_LDW    260                        // 256 + 4-float pad
#define SCORES_LDS_FLOATS (PP_ROWS * D_ + 16 * PQ_LDW + D_)
#define SCORES_LDS_BYTES  (SCORES_LDS_FLOATS * 4)

__global__ __launch_bounds__(256) void scores_tanh_kernel(
    const float* __restrict__ pp, const float* __restrict__ pq,
    const float* __restrict__ vc, float* __restrict__ scores)
{
    extern __shared__ float smem[];
    float* ppS = smem;                        // [64][256]
    float* pqS = smem + PP_ROWS * D_;         // [16][260]
    float* vcS = pqS + 16 * PQ_LDW;           // [256]

    const int b   = blockIdx.z;
    const int p0  = blockIdx.y * PP_ROWS;
    const int q0  = blockIdx.x << 4;
    const int tid = threadIdx.x;

    const float* ppB = pp + ((long)b * TP_ + p0) * D_;
    const float* pqB = pq + ((long)b * TQ_ + q0) * D_;

    vcS[tid] = vc[tid];                       // D_ == blockDim.x == 256

    // stage pq tile: 16 rows x 64 float4, transposed-free (qi-major, padded)
    {
        const float4* src = reinterpret_cast<const float4*>(pqB);
#pragma unroll
        for (int i = tid; i < 16 * (D_ / 4); i += 256) {
            const int r = i >> 6, c = i & 63;
            *reinterpret_cast<float4*>(&pqS[r * PQ_LDW + (c << 2)]) = src[r * (D_ / 4) + c];
        }
    }
    // stage pp tile: 64 rows x 64 float4
    {
        const float4* src = reinterpret_cast<const float4*>(ppB);
#pragma unroll
        for (int i = tid; i < PP_ROWS * (D_ / 4); i += 256) {
            const int r = i >> 6, c = i & 63;
            *reinterpret_cast<float4*>(&ppS[(r << 8) + (c << 2)]) = src[r * (D_ / 4) + c];
        }
    }
    __syncthreads();

    const int qi = tid & 15;
    const int pr = (tid >> 4) << 2;           // 0,4,...,60
    const float* p0r = &ppS[(pr + 0) << 8];
    const float* p1r = &ppS[(pr + 1) << 8];
    const float* p2r = &ppS[(pr + 2) << 8];
    const float* p3r = &ppS[(pr + 3) << 8];
    const float* xqr = &pqS[qi * PQ_LDW];

    float a0 = 0.0f, a1 = 0.0f, a2 = 0.0f, a3 = 0.0f;

    for (int e = 0; e < D_; e += 4) {
        const float4 xq  = *reinterpret_cast<const float4*>(&xqr[e]);
        const float4 vcv = *reinterpret_cast<const float4*>(&vcS[e]);

#define ROW_STEP(PROW, ACC)                                                   \
        {                                                                     \
            const float4 pv = *reinterpret_cast<const float4*>(&PROW[e]);     \
            float x0 = pv.x + xq.x, x1 = pv.y + xq.y;                         \
            float x2 = pv.z + xq.z, x3 = pv.w + xq.w;                         \
            float t0, t1, t2, t3;                                             \
            asm("v_tanh_f32 %0, %4\n\t"                                       \
                "v_tanh_f32 %1, %5\n\t"                                       \
                "v_tanh_f32 %2, %6\n\t"                                       \
                "v_tanh_f32 %3, %7"                                           \
                : "=&v"(t0), "=&v"(t1), "=&v"(t2), "=&v"(t3)                  \
                : "v"(x0), "v"(x1), "v"(x2), "v"(x3));                        \
            ACC = fmaf(vcv.x, t0, ACC);                                       \
            ACC = fmaf(vcv.y, t1, ACC);                                       \
            ACC = fmaf(vcv.z, t2, ACC);                                       \
            ACC = fmaf(vcv.w, t3, ACC);                                       \
        }
        ROW_STEP(p0r, a0)
        ROW_STEP(p1r, a1)
        ROW_STEP(p2r, a2)
        ROW_STEP(p3r, a3)
#undef ROW_STEP
    }

    const long obase = ((long)b * TP_ + p0 + pr) * TQ_ + (q0 + qi);
    scores[obase]            = a0;
    scores[obase + TQ_]      = a1;
    scores[obase + 2 * TQ_]  = a2;
    scores[obase + 3 * TQ_]  = a3;
}

// ---------------------------------------------------------------------------
// In-place softmax over the TP axis of scores[B,TP,TQ]: one wave32 per (b,q)
// column, 16 strided elements per lane, __shfl_xor wave reductions.
// ---------------------------------------------------------------------------
__global__ __launch_bounds__(256) void softmax_tp_kernel(float* __restrict__ s)
{
    const int col  = blockIdx.x * (blockDim.x >> 5) + (threadIdx.x >> 5);
    const int lane = threadIdx.x & 31;
    const int b = col / TQ_;
    const int q = col % TQ_;
    float* base = s + (long)b * TP_ * TQ_ + q;

    float v[TP_ / 32];
    float m = -__builtin_inff();
#pragma unroll
    for (int i = 0; i < TP_ / 32; ++i) {
        v[i] = base[(long)(lane + (i << 5)) * TQ_];
        m = fmaxf(m, v[i]);
    }
#pragma unroll
    for (int o = 16; o > 0; o >>= 1) m = fmaxf(m, __shfl_xor(m, o, 32));

    float sum = 0.0f;
#pragma unroll
    for (int i = 0; i < TP_ / 32; ++i) {
        v[i] = __expf(v[i] - m);
        sum += v[i];
    }
#pragma unroll
    for (int o = 16; o > 0; o >>= 1) sum += __shfl_xor(sum, o, 32);

    const float inv = 1.0f / sum;
#pragma unroll
    for (int i = 0; i < TP_ / 32; ++i)
        base[(long)(lane + (i << 5)) * TQ_] = v[i] * inv;
}

// ---------------------------------------------------------------------------
extern "C" void kernel_launch(void* const* d_in, const int* in_sizes, int n_in,
                              void* d_out, int out_size, void* d_ws, size_t ws_size,
                              hipStream_t stream)
{
    const float* q  = (const float*)d_in[0];   // [B,TQ,D]
    const float* p  = (const float*)d_in[1];   // [B,TP,D]
    const float* W0 = (const float*)d_in[2];   // [D,D]
    const float* W1 = (const float*)d_in[3];   // [D,D]
    const float* vc = (const float*)d_in[4];   // [D,1]
    float* out = (float*)d_out;                // [B,TP,D]

    float* pq_ws     = (float*)d_ws;                                   // B*TQ*D
    float* pp_ws     = pq_ws + (long)B_ * TQ_ * D_;                    // B*TP*D
    float* scores_ws = pp_ws + (long)B_ * TP_ * D_;                    // B*TP*TQ

    // Allow >64KB dynamic LDS for the scores kernel (gfx1250: 320KB/WG max).
    // Host-side attribute set; deterministic and graph-capture safe.
    (void)hipFuncSetAttribute((const void*)scores_tanh_kernel,
                              hipFuncAttributeMaxDynamicSharedMemorySize,
                              SCORES_LDS_BYTES);

    // 1) prod_q = q @ W0 : flatten batch -> M = B*TQ = 2048, N = K = 256
    {
        const int M = B_ * TQ_, N = D_, K = D_;
        const int waves = (M >> 4) * (N >> 4);
        dim3 grid(waves / 8, 1, 1);
        gemm_f32_wmma<<<grid, 256, 0, stream>>>(q, W0, pq_ws, M, N, K, 0, 0, 0);
    }
    // 2) prod_p = p @ W1
    {
        const int M = B_ * TP_, N = D_, K = D_;
        const int waves = (M >> 4) * (N >> 4);
        dim3 grid(waves / 8, 1, 1);
        gemm_f32_wmma<<<grid, 256, 0, stream>>>(p, W1, pp_ws, M, N, K, 0, 0, 0);
    }
    // 3) scores[b,p,q] = sum_e vc[e]*tanh(pp+pq)   (the hot kernel)
    {
        dim3 grid(TQ_ / 16, TP_ / PP_ROWS, B_);
        scores_tanh_kernel<<<grid, 256, SCORES_LDS_BYTES, stream>>>(
            pp_ws, pq_ws, vc, scores_ws);
    }
    // 4) softmax over the TP axis, in place -> weights
    {
        const int cols = B_ * TQ_;
        softmax_tp_kernel<<<cols / 8, 256, 0, stream>>>(scores_ws);
    }
    // 5) out[b] = weights[b] (TPxTQ) @ q[b] (TQxD), batched over z
    {
        const int M = TP_, N = D_, K = TQ_;
        const int waves = (M >> 4) * (N >> 4);     // 512 per batch
        dim3 grid(waves / 8, 1, B_);
        gemm_f32_wmma<<<grid, 256, 0, stream>>>(
            scores_ws, q, out, M, N, K,
            (long)TP_ * TQ_, (long)TQ_ * D_, (long)TP_ * D_);
    }
}